// KeypointsLoss_80934363726256
// MI455X (gfx1250) — compile-verified
//
#include <hip/hip_runtime.h>
#include <math.h>

// Problem constants from the reference: B=32, P=8, K=17, H=W=192, SIGMA=3.
#define NB 32
#define NP 8
#define NK 17
#define NH 192
#define NW 192

typedef __attribute__((ext_vector_type(2))) float v2f;
typedef __attribute__((ext_vector_type(8))) float v8f;

// One block per (b,k). Builds rank-8 target tiles with V_WMMA_F32_16X16X4_F32
// and accumulates sum_{h,w} (pred - target)^2 into partial[b*NK + k].
__global__ __launch_bounds__(256) void kp_partial_kernel(
    const float* __restrict__ pred,       // [B,K,H,W]
    const float* __restrict__ keypoints,  // [B,P,K,2]
    const int*   __restrict__ vis,        // [B,P,K]
    float* __restrict__ partial)          // [B*K]
{
    __shared__ float g_row[NP * NH];   // g_row[p][h], validity folded in
    __shared__ float g_col[NP * NW];   // g_col[p][w]
    __shared__ float xs[NP], ys[NP];
    __shared__ int   vl[NP];
    __shared__ float red[256];

    const int bk  = blockIdx.x;
    const int b   = bk / NK;
    const int k   = bk % NK;
    const int tid = threadIdx.x;

    // Per-person center/validity (reference truncates toward zero).
    if (tid < NP) {
        const int p = tid;
        const float kx = keypoints[(((b * NP + p) * NK + k) * 2) + 0];
        const float ky = keypoints[(((b * NP + p) * NK + k) * 2) + 1];
        const int x = (int)(kx * (float)(NW - 1));   // trunc toward zero
        const int y = (int)(ky * (float)(NH - 1));
        const int v = vis[(b * NP + p) * NK + k];
        vl[p] = (v > 0) && (x >= 0) && (x < NW) && (y >= 0) && (y < NH);
        xs[p] = (float)x;
        ys[p] = (float)y;
    }
    __syncthreads();

    // Separable gaussians into LDS. Note the reference's (row<-x, col<-y) swap.
    const float inv2s2 = 1.0f / (2.0f * 3.0f * 3.0f);
    for (int idx = tid; idx < NP * NH; idx += 256) {
        const int p = idx / NH;
        const int h = idx % NH;
        const float dr = (float)h - xs[p];
        const float dc = (float)h - ys[p];
        g_row[idx] = vl[p] ? __expf(-dr * dr * inv2s2) : 0.0f;
        g_col[idx] = __expf(-dc * dc * inv2s2);
    }
    __syncthreads();

    const int wave = tid >> 5;     // 8 waves per block
    const int lane = tid & 31;
    const int mn   = lane & 15;    // M for A-frag, N for B/C/D frags
    const int half = lane >> 4;    // lane-half: selects K pair (A/B), M+8 (C/D)

    const float* __restrict__ predbk =
        pred + (size_t)(b * NK + k) * (size_t)(NH * NW);

    float acc = 0.0f;

    // 144 tiles of 16x16; 18 contiguous tiles per wave (uniform trip count,
    // no divergence -> EXEC stays all-ones for WMMA).
    for (int i = 0; i < 18; ++i) {
        const int t  = wave * 18 + i;
        const int h0 = (t / 12) * 16;
        const int w0 = (t % 12) * 16;

        v8f c = {0.f, 0.f, 0.f, 0.f, 0.f, 0.f, 0.f, 0.f};

        // Two K=4 steps cover persons 0..7.
        // A (16x4 f32): lane half holds K = 2*half + {0,1}.
        // B (4x16 f32): VGPR0 = rows {K0,K2} across halves, VGPR1 = {K1,K3}.
#pragma unroll
        for (int s = 0; s < 2; ++s) {
            const int pbase = s * 4 + half * 2;
            v2f a, bb;
            a.x  = g_row[(pbase + 0) * NH + h0 + mn];
            a.y  = g_row[(pbase + 1) * NH + h0 + mn];
            bb.x = g_col[(pbase + 0) * NW + w0 + mn];
            bb.y = g_col[(pbase + 1) * NW + w0 + mn];
            c = __builtin_amdgcn_wmma_f32_16x16x4_f32(
                    /*neg_a=*/false, a, /*neg_b=*/false, bb,
                    /*c_mod=*/(short)0, c, /*reuse_a=*/false, /*reuse_b=*/false);
        }

        // C/D layout: VGPR j -> M = j + 8*half, N = lane&15.
#pragma unroll
        for (int j = 0; j < 8; ++j) {
            const int m  = j + 8 * half;
            const float pv = predbk[(h0 + m) * NW + (w0 + mn)];
            const float d  = pv - c[j];
            acc += d * d;
        }
    }

    // Block reduction -> one partial per (b,k).
    red[tid] = acc;
    __syncthreads();
    for (int s = 128; s > 0; s >>= 1) {
        if (tid < s) red[tid] += red[tid + s];
        __syncthreads();
    }
    if (tid == 0) partial[bk] = red[0];
}

// Single wave: lane b (B==32==wave32) normalizes its batch's partial sums by
// denom[b] and a deterministic LDS tree reduction produces the scalar loss.
__global__ __launch_bounds__(32) void kp_final_kernel(
    const float* __restrict__ partial,   // [B*K]
    const int*   __restrict__ vis,       // [B,P,K]
    float* __restrict__ out)             // [1]
{
    __shared__ float red[32];
    const int b = threadIdx.x;

    int dsum = 0;
    for (int i = 0; i < NP * NK; ++i) dsum += vis[b * NP * NK + i];
    float num = 0.0f;
    for (int kk = 0; kk < NK; ++kk) num += partial[b * NK + kk];
    red[b] = num / ((float)dsum + 1e-6f);
    __syncthreads();
    for (int s = 16; s > 0; s >>= 1) {
        if (b < s) red[b] += red[b + s];
        __syncthreads();
    }
    if (b == 0) out[0] = red[0] / (float)NB;
}

extern "C" void kernel_launch(void* const* d_in, const int* in_sizes, int n_in,
                              void* d_out, int out_size, void* d_ws, size_t ws_size,
                              hipStream_t stream) {
    const float* pred = (const float*)d_in[0];   // [32,17,192,192] f32
    const float* kp   = (const float*)d_in[1];   // [32,8,17,2] f32
    const int*   vis  = (const int*)d_in[2];     // [32,8,17] i32
    float* partial    = (float*)d_ws;            // 544 floats of scratch
    float* outp       = (float*)d_out;

    kp_partial_kernel<<<NB * NK, 256, 0, stream>>>(pred, kp, vis, partial);
    kp_final_kernel<<<1, 32, 0, stream>>>(partial, vis, outp);
}